// AffinityLoss_12060268167914
// MI455X (gfx1250) — compile-verified
//
#include <hip/hip_runtime.h>

typedef __attribute__((ext_vector_type(2))) float v2f;
typedef __attribute__((ext_vector_type(8))) float v8f;

#define C_CLASSES 100
#define C_PAD     112          // pad classes to 7 tiles of 16
#define NCT       7            // class tiles
#define D_DIM     256
#define CHUNK     64           // rows staged per iteration
#define FSTRIDE   260          // padded LDS row stride (floats): breaks bank conflicts,
                               // keeps 16B alignment (260*4 = 1040 = 65*16)
#define NTHREADS  512
#define NWAVES    16
#define SEGS      8            // 8 x B128 async segments per thread per chunk
                               // 512 thr * 8 * 16B = 64KB = CHUNK*D_DIM*4  (exact)

// ---------------------------------------------------------------------------
// Workspace layout (d_ws):
//   [0]   double sumsq
//   [2..] float  g_sums[C_PAD * D_DIM]
//   [..]  float  g_counts[C_PAD]
// ---------------------------------------------------------------------------

__global__ __launch_bounds__(256) void affinity_zero(float* p, int n) {
    int i = blockIdx.x * blockDim.x + threadIdx.x;
    if (i < n) p[i] = 0.0f;
}

// DMA one 64x256 f32 chunk into an LDS buffer (ASYNCcnt path, no VGPR staging).
__device__ __forceinline__ void async_stage_chunk(const float* __restrict__ gbase,
                                                  float* ldsbuf, int tid) {
#pragma unroll
    for (int i = 0; i < SEGS; ++i) {
        int seg  = tid + i * NTHREADS;          // 16B segment id, 0..4095
        int row  = seg >> 6;                    // 64 segments per 256-float row
        int col4 = seg & 63;
        float*   lptr = ldsbuf + row * FSTRIDE + col4 * 4;
        unsigned lds  = (unsigned)(unsigned long long)(void*)lptr;  // LDS byte offset
        unsigned goff = (unsigned)seg * 16u;                        // global byte offset
        asm volatile("global_load_async_to_lds_b128 %0, %1, %2"
                     :: "v"(lds), "v"(goff), "s"(gbase) : "memory");
    }
}

// argmax over 100 scores for 64 rows (threads 0..63), float4-vectorized.
__device__ __forceinline__ void do_argmax(const float* __restrict__ labels,
                                          long rowBase, int tid,
                                          int* lblbuf, float* ldsCnt) {
    if (tid < CHUNK) {
        const float4* lr = (const float4*)(labels + (rowBase + tid) * C_CLASSES);
        float best = -INFINITY;
        int bi = 0, idx = 0;
        for (int j = 0; j < C_CLASSES / 4; ++j) {
            float4 v = lr[j];
            if (v.x > best) { best = v.x; bi = idx; } idx++;
            if (v.y > best) { best = v.y; bi = idx; } idx++;
            if (v.z > best) { best = v.z; bi = idx; } idx++;
            if (v.w > best) { best = v.w; bi = idx; } idx++;
        }
        lblbuf[tid] = bi;
        atomicAdd(&ldsCnt[bi], 1.0f);           // ds_add_f32
    }
}

// Fused pass: async double-buffered staging + argmax + one-hot fp32 WMMA GEMM.
__global__ __launch_bounds__(NTHREADS) void affinity_main(
    const float* __restrict__ features, const float* __restrict__ labels,
    float* __restrict__ g_sums, float* __restrict__ g_counts,
    double* __restrict__ g_sumsq, int N)
{
    __shared__ float fbuf[2][CHUNK * FSTRIDE];  // double-buffered feature chunks
    __shared__ int   lbl[2][CHUNK];
    __shared__ float ldsCnt[C_PAD];
    __shared__ float ldsRed[NWAVES];

    const int tid  = threadIdx.x;
    const int lane = tid & 31;
    const int wave = tid >> 5;                  // D-tile owned by this wave

    for (int i = tid; i < C_PAD; i += NTHREADS) ldsCnt[i] = 0.0f;
    __syncthreads();                            // counts zeroed before any atomicAdd

    v8f acc[NCT];
#pragma unroll
    for (int ct = 0; ct < NCT; ++ct) acc[ct] = (v8f){};

    float sumsq = 0.0f;

    const int nChunks = N / CHUNK;
    const int stride  = gridDim.x;
    const int dbase = wave * 16 + (lane & 15);  // this lane's D column
    const int rsel0 = (lane < 16) ? 0 : 2;      // A/B K-row mapping (ISA 7.12.2)
    const int rsel1 = rsel0 + 1;

    // Prologue: stage + label first chunk.
    int cb = 0;
    {
        int c0 = blockIdx.x;
        if (c0 < nChunks) {
            async_stage_chunk(features + (long)c0 * CHUNK * D_DIM, &fbuf[0][0], tid);
            do_argmax(labels, (long)c0 * CHUNK, tid, lbl[0], ldsCnt);
        }
    }

    for (int ch = blockIdx.x; ch < nChunks; ch += stride) {
        const int  nx = ch + stride;
        const bool hn = nx < nChunks;

        __syncthreads();   // all waves done reading fbuf[cb^1]/lbl[cb^1] -> reusable

        if (hn) {
            // Overlap: DMA + argmax for NEXT chunk while current one is consumed.
            async_stage_chunk(features + (long)nx * CHUNK * D_DIM, &fbuf[cb ^ 1][0], tid);
            do_argmax(labels, (long)nx * CHUNK, tid, lbl[cb ^ 1], ldsCnt);
            long nn = (long)(nx + stride) * CHUNK;          // labels 2 chunks ahead
            if (nn < (long)N && tid < CHUNK)
                __builtin_prefetch(labels + (nn + tid) * C_CLASSES, 0, 1);
            asm volatile("s_wait_asynccnt 8" ::: "memory"); // current chunk landed
        } else {
            asm volatile("s_wait_asynccnt 0" ::: "memory"); // drain
        }
        __syncthreads();   // all waves' DMA for fbuf[cb] visible

        // One-hot GEMM: sums[112x256] += OneHot(64x112)^T * F(64x256),
        // K in steps of 4 with V_WMMA_F32_16X16X4_F32 (exact fp32).
        const float* fb = &fbuf[cb][0];
        const int*   lb = &lbl[cb][0];
        for (int k = 0; k < CHUNK / 4; ++k) {
            const int r = k * 4;
            v2f b;                                          // B: 4x16 feature slab
            b.x = fb[(r + rsel0) * FSTRIDE + dbase];
            b.y = fb[(r + rsel1) * FSTRIDE + dbase];
            sumsq += b.x * b.x + b.y * b.y;                 // each element hit once
            const int l0 = lb[r + rsel0];
            const int l1 = lb[r + rsel1];
#pragma unroll
            for (int ct = 0; ct < NCT; ++ct) {
                const int c = ct * 16 + (lane & 15);
                v2f a;                                      // A: 16x4 one-hot tile
                a.x = (l0 == c) ? 1.0f : 0.0f;
                a.y = (l1 == c) ? 1.0f : 0.0f;
                acc[ct] = __builtin_amdgcn_wmma_f32_16x16x4_f32(
                    false, a, false, b, (short)0, acc[ct], false, false);
            }
        }
        cb ^= 1;
    }

    // Flush per-class sums: C/D tile layout = lane -> N, VGPR v -> M.
#pragma unroll
    for (int ct = 0; ct < NCT; ++ct) {
#pragma unroll
        for (int v = 0; v < 8; ++v) {
            int cls = ct * 16 + v + ((lane < 16) ? 0 : 8);
            atomicAdd(&g_sums[cls * D_DIM + dbase], acc[ct][v]);
        }
    }

    __syncthreads();
    for (int i = tid; i < C_PAD; i += NTHREADS) atomicAdd(&g_counts[i], ldsCnt[i]);

    // sumsq: wave shuffle reduce -> LDS -> one f64 atomic per workgroup.
    for (int off = 16; off > 0; off >>= 1) sumsq += __shfl_down(sumsq, off, 32);
    if (lane == 0) ldsRed[wave] = sumsq;
    __syncthreads();
    if (wave == 0) {
        float s = (lane < NWAVES) ? ldsRed[lane] : 0.0f;
        for (int off = 8; off > 0; off >>= 1) s += __shfl_down(s, off, 32);
        if (lane == 0) atomicAdd(g_sumsq, (double)s);
    }
}

// Finishing kernel: O(C*D) math, one workgroup. Thread t owns column d = t.
__global__ __launch_bounds__(D_DIM) void affinity_finish(
    const float* __restrict__ g_sums, const float* __restrict__ g_counts,
    const double* __restrict__ g_sumsq, float* __restrict__ out)
{
    __shared__ float cnt[C_CLASSES];
    __shared__ float redT1[D_DIM];
    __shared__ float redT2[D_DIM];
    __shared__ float redIV[D_DIM];

    const int t = threadIdx.x;
    if (t < C_CLASSES) cnt[t] = g_counts[t];
    __syncthreads();

    float colsum = 0.0f, T1 = 0.0f, T2 = 0.0f;
    for (int c = 0; c < C_CLASSES; ++c) {
        float n = cnt[c];
        float s = g_sums[c * D_DIM + t];
        float m = (n > 0.0f) ? (s / fmaxf(n, 1.0f)) : 0.0f;
        float ctr = m + 1e-6f;
        colsum += ctr;
        T1 += ctr * s;           // sum_c ctr_c . s_c
        T2 += n * ctr * ctr;     // sum_c n_c ||ctr_c||^2
    }
    const float mean_t = colsum / (float)C_CLASSES;

    float iv = 0.0f;
    for (int c = 0; c < C_CLASSES; ++c) {
        float n = cnt[c];
        float s = g_sums[c * D_DIM + t];
        float m = (n > 0.0f) ? (s / fmaxf(n, 1.0f)) : 0.0f;
        float ctr = m + 1e-6f;
        float d = ctr - mean_t;
        iv += d * d;
    }

    redT1[t] = T1; redT2[t] = T2; redIV[t] = iv;
    __syncthreads();
    for (int off = D_DIM / 2; off > 0; off >>= 1) {
        if (t < off) {
            redT1[t] += redT1[t + off];
            redT2[t] += redT2[t + off];
            redIV[t] += redIV[t + off];
        }
        __syncthreads();
    }

    if (t == 0) {
        double sumsq = *g_sumsq;
        double intra = sumsq - 2.0 * (double)redT1[0] + (double)redT2[0];
        double inter = (double)redIV[0] / (double)C_CLASSES;
        out[0] = (float)(intra / (inter + 1e-6));
    }
}

extern "C" void kernel_launch(void* const* d_in, const int* in_sizes, int n_in,
                              void* d_out, int out_size, void* d_ws, size_t ws_size,
                              hipStream_t stream) {
    const float* features = (const float*)d_in[0];
    const float* labels   = (const float*)d_in[1];
    const int N = in_sizes[0] / D_DIM;     // 262144

    double* g_sumsq  = (double*)d_ws;
    float*  g_sums   = (float*)d_ws + 2;
    float*  g_counts = g_sums + C_PAD * D_DIM;

    const int zeroN = 2 + C_PAD * D_DIM + C_PAD;
    affinity_zero<<<(zeroN + 255) / 256, 256, 0, stream>>>((float*)d_ws, zeroN);

    affinity_main<<<256, NTHREADS, 0, stream>>>(
        features, labels, g_sums, g_counts, g_sumsq, N);

    affinity_finish<<<1, D_DIM, 0, stream>>>(g_sums, g_counts, g_sumsq, (float*)d_out);
}